// Peak_37211596653114
// MI455X (gfx1250) — compile-verified
//
#include <hip/hip_runtime.h>
#include <math.h>

// ---------------------------------------------------------------------------
// Peaking-EQ biquad (DF2T) over (16, 2, 2^20) f32 via chunked linear scan.
//   Phase 1: per-chunk zero-init final state          (reads x once)
//   Phase 2: per-sequence scan of chunk states (M^L)  (tiny)
//   Phase 3: block state-space form, 16 samples x 16 chunks per wave,
//            Y = H*X + G*S0 and F*X via V_WMMA_F32_16X16X4_F32.
// ---------------------------------------------------------------------------

typedef __attribute__((ext_vector_type(2))) float v2f;
typedef __attribute__((ext_vector_type(4))) float f4;
typedef __attribute__((ext_vector_type(8))) float v8f;

#define TLEN      1048576     // samples per sequence
#define NSEQ      32          // 16 batch * 2 channels
#define LCHUNK    1024        // samples per chunk
#define NCHUNK    1024        // chunks per sequence (TLEN / LCHUNK)
#define NCHTOT    (NSEQ * NCHUNK)   // 32768 total chunks
#define LOG2_L    10

#define MIN_F 33.0f
#define MAX_F 17500.0f
#define MIN_Q 0.2f
#define MAX_Q 20.0f
#define PI_F  3.14159265358979323846f

struct Coef { float b0, k1, k2, a1, a2; };

__device__ __forceinline__ Coef make_coeffs(float fr, float qr, float g) {
    float freq  = (MAX_F - MIN_F) / (1.0f + __expf(-fr)) + MIN_F;
    float Q     = (MAX_Q - MIN_Q) / (1.0f + __expf(-qr)) + MIN_Q;
    float w0    = 2.0f * PI_F * freq / 44100.0f;
    float A     = powf(10.0f, g * (1.0f / 40.0f));
    float sw, cw;
    __sincosf(w0, &sw, &cw);
    float alpha = sw / (2.0f * Q);
    float b0 = 1.0f + alpha * A;
    float b1 = -2.0f * cw;
    float b2 = 1.0f - alpha * A;
    float a0 = 1.0f + alpha / A;
    float a2 = 1.0f - alpha / A;
    float inv = 1.0f / a0;
    b0 *= inv; b1 *= inv; b2 *= inv;
    float a1 = b1;          // already normalized (a1 == b1 pre-normalization)
    a2 *= inv;
    Coef c;
    c.b0 = b0; c.a1 = a1; c.a2 = a2;
    // state recurrence: s1' = k1*x - a1*s1 + s2 ; s2' = k2*x - a2*s1 ; y = b0*x + s1
    c.k1 = b1 - a1 * b0;
    c.k2 = b2 - a2 * b0;
    return c;
}

// ------------------------- Phase 1: chunk local states ----------------------
__global__ void __launch_bounds__(256)
peak_phase1(const float* __restrict__ x,
            const float* __restrict__ fr, const float* __restrict__ qr,
            const float* __restrict__ gn, float* __restrict__ cs) {
    Coef c = make_coeffs(*fr, *qr, *gn);
    int gid = blockIdx.x * blockDim.x + threadIdx.x;       // one thread / chunk
    const f4* px = (const f4*)(x + (size_t)gid * LCHUNK);
    float s1 = 0.0f, s2 = 0.0f;
#define P1STEP(xx) { float t1 = fmaf(c.k1, (xx), fmaf(-c.a1, s1, s2)); \
                     float t2 = fmaf(c.k2, (xx), -c.a2 * s1);          \
                     s1 = t1; s2 = t2; }
    for (int i = 0; i < LCHUNK / 4; ++i) {
        f4 v = px[i];
        P1STEP(v.x) P1STEP(v.y) P1STEP(v.z) P1STEP(v.w)
    }
#undef P1STEP
    cs[2 * gid + 0] = s1;
    cs[2 * gid + 1] = s2;
}

// ------------------- Phase 2: inter-chunk scan (per sequence) ---------------
__global__ void peak_phase2(const float* __restrict__ fr, const float* __restrict__ qr,
                            const float* __restrict__ gn,
                            const float* __restrict__ cs, float* __restrict__ is) {
    Coef c = make_coeffs(*fr, *qr, *gn);
    int seq = threadIdx.x;                 // 32 threads, one per sequence
    if (seq >= NSEQ) return;
    // M = [[-a1, 1], [-a2, 0]];  M^LCHUNK via repeated squaring
    float m00 = -c.a1, m01 = 1.0f, m10 = -c.a2, m11 = 0.0f;
    for (int i = 0; i < LOG2_L; ++i) {
        float n00 = m00 * m00 + m01 * m10;
        float n01 = m00 * m01 + m01 * m11;
        float n10 = m10 * m00 + m11 * m10;
        float n11 = m10 * m01 + m11 * m11;
        m00 = n00; m01 = n01; m10 = n10; m11 = n11;
    }
    const float* p = cs + (size_t)seq * NCHUNK * 2;
    float*       q = is + (size_t)seq * NCHUNK * 2;
    float s1 = 0.0f, s2 = 0.0f;
    for (int k = 0; k < NCHUNK; ++k) {
        q[2 * k + 0] = s1;
        q[2 * k + 1] = s2;
        float c1 = p[2 * k + 0], c2 = p[2 * k + 1];
        float t1 = m00 * s1 + m01 * s2 + c1;
        float t2 = m10 * s1 + m11 * s2 + c2;
        s1 = t1; s2 = t2;
    }
}

// -------- Phase 3: outputs via 16x16 block state-space form + WMMA ----------
// One wave owns 16 consecutive chunks (columns). Per 16-sample block:
//   Y(16x16)  = G(16x2)*S0(2x16) + H(16x16)*X(16x16)   -> 5 wmma f32_16x16x4
//   FX(2x16)  = F(2x16)*X(16x16)                       -> 4 wmma f32_16x16x4
//   S0        = M^16*S0 + FX
__global__ void __launch_bounds__(256)
peak_phase3(const float* __restrict__ x,
            const float* __restrict__ fr, const float* __restrict__ qr,
            const float* __restrict__ gn,
            const float* __restrict__ is, float* __restrict__ out) {
    __shared__ float sh[16 * 5 + 4];   // h[16] g1[16] g2[16] f1[16] f2[16] M16[4]
    Coef c = make_coeffs(*fr, *qr, *gn);

    if (threadIdx.x == 0) {
        float m00 = -c.a1, m01 = 1.0f, m10 = -c.a2, m11 = 0.0f;
        // impulse response h[d]: h[0]=b0, h[d]=e1^T M^(d-1) c
        sh[0] = c.b0;
        float v0 = c.k1, v1 = c.k2;
        for (int d = 1; d < 16; ++d) {
            sh[d] = v0;
            float n0 = m00 * v0 + m01 * v1, n1 = m10 * v0 + m11 * v1;
            v0 = n0; v1 = n1;
        }
        // G columns: g_i[t] = e1^T M^t e_i
        float u0 = 1.0f, u1 = 0.0f;
        for (int t = 0; t < 16; ++t) {
            sh[16 + t] = u0;
            float n0 = m00 * u0 + m01 * u1, n1 = m10 * u0 + m11 * u1;
            u0 = n0; u1 = n1;
        }
        u0 = 0.0f; u1 = 1.0f;
        for (int t = 0; t < 16; ++t) {
            sh[32 + t] = u0;
            float n0 = m00 * u0 + m01 * u1, n1 = m10 * u0 + m11 * u1;
            u0 = n0; u1 = n1;
        }
        // F columns: F[:,t] = M^(15-t) c  (t = 0..15)
        float w0 = c.k1, w1 = c.k2;
        for (int t = 15; t >= 0; --t) {
            sh[48 + t] = w0;
            sh[64 + t] = w1;
            float n0 = m00 * w0 + m01 * w1, n1 = m10 * w0 + m11 * w1;
            w0 = n0; w1 = n1;
        }
        // M^16: square 4 times
        float q00 = m00, q01 = m01, q10 = m10, q11 = m11;
        for (int i = 0; i < 4; ++i) {
            float n00 = q00 * q00 + q01 * q10;
            float n01 = q00 * q01 + q01 * q11;
            float n10 = q10 * q00 + q11 * q10;
            float n11 = q10 * q01 + q11 * q11;
            q00 = n00; q01 = n01; q10 = n10; q11 = n11;
        }
        sh[80] = q00; sh[81] = q01; sh[82] = q10; sh[83] = q11;
    }
    __syncthreads();

    int lane = threadIdx.x & 31;
    int m    = lane & 15;          // column (chunk) index within wave / A row
    int hi   = lane >> 4;          // half-wave select
    int wave = blockIdx.x * (blockDim.x >> 5) + (threadIdx.x >> 5); // 0..2047

    size_t mychunk = (size_t)wave * 16 + (size_t)m;   // chunks tile the flat array
    const float* px = x   + mychunk * LCHUNK;
    float*       py = out + mychunk * LCHUNK;

    // A operands (loop-invariant), per-lane from LDS tables.
    // 32-bit A 16x4 layout: lanes 0-15 row m hold K={0,1}; lanes 16-31 K={2,3}.
    v2f aH[4], aF[4], aG;
#pragma unroll
    for (int s = 0; s < 4; ++s) {
        int k0 = 4 * s + 2 * hi;
        int k1i = k0 + 1;
        float hx = (m >= k0)  ? sh[m - k0]  : 0.0f;
        float hy = (m >= k1i) ? sh[m - k1i] : 0.0f;
        aH[s] = (v2f){hx, hy};
        float fx = 0.0f, fy = 0.0f;
        if (m == 0) { fx = sh[48 + k0]; fy = sh[48 + k1i]; }
        if (m == 1) { fx = sh[64 + k0]; fy = sh[64 + k1i]; }
        aF[s] = (v2f){fx, fy};
    }
    aG = hi ? (v2f){0.0f, 0.0f} : (v2f){sh[16 + m], sh[32 + m]};

    float m16_00 = sh[80], m16_01 = sh[81], m16_10 = sh[82], m16_11 = sh[83];

    float s1 = hi ? 0.0f : is[2 * mychunk + 0];
    float s2 = hi ? 0.0f : is[2 * mychunk + 1];

    for (int b = 0; b < LCHUNK / 16; ++b) {
        // X block: lane (m,hi) loads column m rows {4s+2*hi, 4s+2*hi+1} as v2f
        const float* pb = px + 16 * b + 2 * hi;
        v2f r0 = *(const v2f*)(pb + 0);
        v2f r1 = *(const v2f*)(pb + 4);
        v2f r2 = *(const v2f*)(pb + 8);
        v2f r3 = *(const v2f*)(pb + 12);

        // B operand for G*S0: rows {0,1}=state (lanes<16), rows {2,3}=0
        v2f bS0 = hi ? (v2f){0.0f, 0.0f} : (v2f){s1, s2};

        v8f acc = {};
        acc = __builtin_amdgcn_wmma_f32_16x16x4_f32(false, aG,    false, bS0, (short)0, acc, false, false);
        acc = __builtin_amdgcn_wmma_f32_16x16x4_f32(false, aH[0], false, r0,  (short)0, acc, false, false);
        acc = __builtin_amdgcn_wmma_f32_16x16x4_f32(false, aH[1], false, r1,  (short)0, acc, false, false);
        acc = __builtin_amdgcn_wmma_f32_16x16x4_f32(false, aH[2], false, r2,  (short)0, acc, false, false);
        acc = __builtin_amdgcn_wmma_f32_16x16x4_f32(false, aH[3], false, r3,  (short)0, acc, false, false);

        v8f fxacc = {};
        fxacc = __builtin_amdgcn_wmma_f32_16x16x4_f32(false, aF[0], false, r0, (short)0, fxacc, false, false);
        fxacc = __builtin_amdgcn_wmma_f32_16x16x4_f32(false, aF[1], false, r1, (short)0, fxacc, false, false);
        fxacc = __builtin_amdgcn_wmma_f32_16x16x4_f32(false, aF[2], false, r2, (short)0, fxacc, false, false);
        fxacc = __builtin_amdgcn_wmma_f32_16x16x4_f32(false, aF[3], false, r3, (short)0, fxacc, false, false);

        // state update (valid in lanes 0-15; D rows 0,1 live in lanes 0-15)
        float ns1 = m16_00 * s1 + m16_01 * s2 + fxacc[0];
        float ns2 = m16_10 * s1 + m16_11 * s2 + fxacc[1];
        s1 = hi ? 0.0f : ns1;
        s2 = hi ? 0.0f : ns2;

        // store Y: VGPR v holds row v (lanes<16) / row v+8 (lanes>=16)
        float* po = py + 16 * b + 8 * hi;
        f4 o0 = (f4){acc[0], acc[1], acc[2], acc[3]};
        f4 o1 = (f4){acc[4], acc[5], acc[6], acc[7]};
        *(f4*)(po + 0) = o0;
        *(f4*)(po + 4) = o1;
    }
}

// ---------------------------------------------------------------------------
extern "C" void kernel_launch(void* const* d_in, const int* in_sizes, int n_in,
                              void* d_out, int out_size, void* d_ws, size_t ws_size,
                              hipStream_t stream) {
    const float* x  = (const float*)d_in[0];
    const float* fr = (const float*)d_in[1];
    const float* qr = (const float*)d_in[2];
    const float* gn = (const float*)d_in[3];
    float* out = (float*)d_out;

    float* cs = (float*)d_ws;                    // per-chunk zero-init final states
    float* is = cs + (size_t)NCHTOT * 2;         // per-chunk initial states
    (void)in_sizes; (void)n_in; (void)out_size; (void)ws_size;

    // Phase 1: 32768 chunks, 1 thread each
    peak_phase1<<<NCHTOT / 256, 256, 0, stream>>>(x, fr, qr, gn, cs);
    // Phase 2: 1 wave, one lane per sequence
    peak_phase2<<<1, 32, 0, stream>>>(fr, qr, gn, cs, is);
    // Phase 3: 2048 waves (16 chunks each), 8 waves per block
    peak_phase3<<<(NCHTOT / 16) / 8, 256, 0, stream>>>(x, fr, qr, gn, is, out);
}